// _LinearShapeletTransform_61460982006082
// MI455X (gfx1250) — compile-verified
//
#include <hip/hip_runtime.h>
#include <math.h>

// Problem constants (match reference)
#define B_    32
#define T_    1000
#define C_    6
#define S_    20
#define NSS_  25
#define LPER_ 3
#define P_    100
#define K_    60     // S * LPER
#define OUT_  10
#define EPS_  1e-5f

typedef __attribute__((ext_vector_type(2))) float v2f;
typedef __attribute__((ext_vector_type(8))) float v8f;

// ---------------------------------------------------------------------------
// Kernel 1: continuous shapelet transform.
// One wave32 per (b,k); lanes stride over the P start positions; LDS-staged
// `times` for the binary search; wave shuffle min-reduction.
// ---------------------------------------------------------------------------
__global__ __launch_bounds__(128) void shapelet_sim_kernel(
    const float* __restrict__ times,      // [T]
    const float* __restrict__ X,          // [B,T,C]
    const float* __restrict__ shapelets,  // [S,NSS,C]
    const float* __restrict__ lengths,    // [K]
    float* __restrict__ sim)              // [B,K]
{
    __shared__ float s_times[T_];
    for (int i = threadIdx.x; i < T_; i += blockDim.x) s_times[i] = times[i];
    __syncthreads();

    const int wave = threadIdx.x >> 5;         // 0..3
    const int lane = threadIdx.x & 31;
    const int bk   = blockIdx.x * 4 + wave;    // flat (b,k)
    const int b    = bk / K_;
    const int k    = bk % K_;
    const int s    = k / LPER_;

    const float len      = lengths[k];
    const float t0       = s_times[0];
    const float t1       = s_times[T_ - 1];
    const float span     = t1 - len - t0;                 // (t1 - length - t0)
    const float posstep  = 1.0f / (float)(P_ - 1);
    const float fracstep = 1.0f / (float)(NSS_ - 1);

    const float* __restrict__ shp = shapelets + s * (NSS_ * C_);
    const float* __restrict__ Xb  = X + (size_t)b * (T_ * C_);

    float best = 3.0e38f;  // min over this lane's start positions (pre-sqrt)

    for (int p = lane; p < P_; p += 32) {
        const float start = t0 + (float)p * posstep * span;
        float fprev[C_];
        float seg = 0.0f;

        for (int j = 0; j < NSS_; ++j) {
            const float et = start + (float)j * fracstep * len;

            // searchsorted(times, et, side='right') - 1, clipped to [0, T-2]
            int lo = 0, hi = T_;
            while (lo < hi) {
                const int mid = (lo + hi) >> 1;
                if (s_times[mid] <= et) lo = mid + 1; else hi = mid;
            }
            int idx = lo - 1;
            idx = idx < 0 ? 0 : (idx > T_ - 2 ? T_ - 2 : idx);

            const float tl = s_times[idx];
            const float tr = s_times[idx + 1];
            const float w  = (et - tl) / (tr - tl);

            const float* __restrict__ xl = Xb + idx * C_;
            const float* __restrict__ sj = shp + j * C_;

            #pragma unroll
            for (int c = 0; c < C_; ++c) {
                const float a   = xl[c];
                const float val = a + w * (xl[c + C_] - a);   // lerp to idx+1
                const float f   = val - sj[c];
                if (j > 0) seg += fprev[c] * fprev[c] + fprev[c] * f + f * f;
                fprev[c] = f;
            }
        }
        best = fminf(best, seg);
    }

    // wave32 min-reduction
    #pragma unroll
    for (int off = 16; off >= 1; off >>= 1)
        best = fminf(best, __shfl_xor(best, off, 32));

    if (lane == 0) {
        const float dt3   = len * fracstep * (1.0f / 3.0f);  // (len/(NSS-1))/3
        const float integ = best * dt3;
        sim[b * K_ + k]   = sqrtf(fmaxf(integ, 1e-12f));
    }
}

// ---------------------------------------------------------------------------
// Kernel 2: out[32,10] = log(sim+EPS)[32,60] @ W.T[60,10] + bias
// via V_WMMA_F32_16X16X4_F32: two 16x16 M-tiles (one per wave), 15 K=4 steps.
// A layout: lane holds M=lane%16; K pair {0,1} (lanes 0-15) / {2,3} (16-31).
// B layout: lane holds N=lane%16; same K split.  D: M = vgpr + 8*(lane>=16).
// ---------------------------------------------------------------------------
__global__ __launch_bounds__(64) void head_wmma_kernel(
    const float* __restrict__ sim,   // [B,K]
    const float* __restrict__ Wm,    // [OUT,K]
    const float* __restrict__ bias,  // [OUT]
    float* __restrict__ out)         // [B,OUT]
{
    const int wave = threadIdx.x >> 5;       // M-tile: rows 0-15 or 16-31
    const int lane = threadIdx.x & 31;
    const int n    = lane & 15;              // column (N) this lane holds
    const int koff = (lane >> 4) << 1;       // 0 or 2
    const int m    = wave * 16 + n;          // row this lane fetches A from

    v8f acc = {};

    for (int kk = 0; kk < K_; kk += 4) {
        const int ka = kk + koff;

        v2f a, bm;
        a.x = logf(sim[m * K_ + ka]     + EPS_);
        a.y = logf(sim[m * K_ + ka + 1] + EPS_);
        // B = W^T (K x N); pad N=10..15 with zeros (uniform select, no branch)
        bm.x = (n < OUT_) ? Wm[n * K_ + ka]     : 0.0f;
        bm.y = (n < OUT_) ? Wm[n * K_ + ka + 1] : 0.0f;

        acc = __builtin_amdgcn_wmma_f32_16x16x4_f32(
            /*neg_a=*/false, a, /*neg_b=*/false, bm,
            /*c_mod=*/(short)0, acc, /*reuse_a=*/false, /*reuse_b=*/false);
    }

    const int mhi = (lane >= 16) ? 8 : 0;
    #pragma unroll
    for (int v = 0; v < 8; ++v) {
        const int mrow = wave * 16 + v + mhi;
        if (n < OUT_) out[mrow * OUT_ + n] = acc[v] + bias[n];
    }
}

// ---------------------------------------------------------------------------
extern "C" void kernel_launch(void* const* d_in, const int* in_sizes, int n_in,
                              void* d_out, int out_size, void* d_ws, size_t ws_size,
                              hipStream_t stream) {
    const float* times     = (const float*)d_in[0];
    const float* X         = (const float*)d_in[1];
    const float* shapelets = (const float*)d_in[2];
    const float* lengths   = (const float*)d_in[3];
    const float* Wm        = (const float*)d_in[4];
    const float* bias      = (const float*)d_in[5];

    float* outp = (float*)d_out;            // [B,OUT] = 320 floats
    float* simp = outp + B_ * OUT_;         // [B,K]   = 1920 floats

    const int nblocks = (B_ * K_) / 4;      // 4 waves per 128-thread block
    shapelet_sim_kernel<<<nblocks, 128, 0, stream>>>(times, X, shapelets, lengths, simp);
    head_wmma_kernel<<<1, 64, 0, stream>>>(simp, Wm, bias, outp);
}